// Sparsemax_51917564674410
// MI455X (gfx1250) — compile-verified
//
#include <hip/hip_runtime.h>
#include <hip/hip_bf16.h>
#include <stdint.h>

// Sparsemax over rows of D=256 fp32, B=131072 rows.
// Memory-bound: 256 MB total traffic -> ~11us floor at 23.3 TB/s.
// Exact sort-free algorithm (Michelot simplex projection), one wave32 per row,
// 8 elements/lane in registers, async global->LDS double-buffered staging.
// Streaming-once data: non-temporal hints on both load and store sides so the
// 256 MB stream does not churn the 192 MB global L2.

#define SPX_D 256
#define WAVES_PER_BLOCK 8
#define ROW_BYTES (SPX_D * 4)

typedef float v4f __attribute__((ext_vector_type(4)));

__device__ __forceinline__ void issue_async_row(const float* __restrict__ in,
                                                int row, uint32_t lds_off_base,
                                                int lane) {
  // Each lane moves two 16B chunks: bytes [lane*16, +16) and [512+lane*16, +16).
  // INST_OFFSET applies to both LDS and global addresses (CDNA5 ISA),
  // so one 64-bit address VGPR pair covers both halves.
  uint32_t lds_off = lds_off_base + (uint32_t)(lane * 16);
  unsigned long long ga =
      (unsigned long long)((const char*)in + (size_t)row * ROW_BYTES + (size_t)(lane * 16));
  asm volatile("global_load_async_to_lds_b128 %0, %1, off th:TH_LOAD_NT"
               :: "v"(lds_off), "v"(ga) : "memory");
  asm volatile("global_load_async_to_lds_b128 %0, %1, off offset:512 th:TH_LOAD_NT"
               :: "v"(lds_off), "v"(ga) : "memory");
}

__global__ __launch_bounds__(WAVES_PER_BLOCK * 32)
void sparsemax_michelot_kernel(const float* __restrict__ in,
                               float* __restrict__ out,
                               int nrows, int nwaves) {
  __shared__ __align__(16) float lds[2 * WAVES_PER_BLOCK * SPX_D];

  const int lane = threadIdx.x & 31;
  const int wave = threadIdx.x >> 5;
  const int gwave = blockIdx.x * WAVES_PER_BLOCK + wave;

  // Low 32 bits of a generic pointer to LDS = byte offset within the
  // workgroup's LDS allocation (aperture form: {SHARED_BASE, offset}).
  const uint32_t wave_lds_byte =
      (uint32_t)(uintptr_t)(&lds[wave * 2 * SPX_D]);

  int row = gwave;
  if (row < nrows) {
    issue_async_row(in, row, wave_lds_byte, lane);  // prologue: fill buf 0
  }

  int buf = 0;
  for (; row < nrows; row += nwaves) {
    const int next = row + nwaves;
    if (next < nrows) {
      // Prefetch next row into the other buffer, then wait only until the
      // *current* row's two transfers are done (ASYNCcnt completes in order:
      // <=2 outstanding leaves exactly the next row's pair in flight).
      issue_async_row(in, next, wave_lds_byte + (uint32_t)((buf ^ 1) * ROW_BYTES), lane);
      asm volatile("s_wait_asynccnt 2" ::: "memory");
    } else {
      asm volatile("s_wait_asynccnt 0" ::: "memory");
    }

    // Pull this lane's 8 contiguous elements from LDS into registers.
    const v4f* p = reinterpret_cast<const v4f*>(
        &lds[wave * 2 * SPX_D + buf * SPX_D + lane * 8]);
    v4f v0 = p[0];
    v4f v1 = p[1];
    float z[8] = {v0.x, v0.y, v0.z, v0.w, v1.x, v1.y, v1.z, v1.w};

    // ---- Michelot projection: active set shrinks monotonically; stop when
    // the active count is stable (set monotone => count-stable <=> set-stable).

    // Iteration 0 hoisted: everything is active, count is exactly SPX_D.
    float s = 0.0f;
#pragma unroll
    for (int e = 0; e < 8; ++e) s += z[e];
#pragma unroll
    for (int m = 16; m >= 1; m >>= 1) s += __shfl_xor(s, m, 32);
    float tau = (s - 1.0f) * (1.0f / (float)SPX_D);

    bool act[8];
#pragma unroll
    for (int e = 0; e < 8; ++e) act[e] = z[e] > tau;

    int prev_n = SPX_D;
    for (int it = 1; it < SPX_D; ++it) {   // hard bound guarantees termination
      float sa = 0.0f;
      int c = 0;
#pragma unroll
      for (int e = 0; e < 8; ++e) {
        sa += act[e] ? z[e] : 0.0f;
        c += act[e] ? 1 : 0;
      }
      // wave32 reduction (two scalars)
#pragma unroll
      for (int m = 16; m >= 1; m >>= 1) {
        sa += __shfl_xor(sa, m, 32);
        c += __shfl_xor(c, m, 32);
      }
      tau = (sa - 1.0f) / (float)c;
      if (c == prev_n) break;              // active set unchanged -> converged
      prev_n = c;
#pragma unroll
      for (int e = 0; e < 8; ++e) act[e] = z[e] > tau;
    }

    v4f o0, o1;
    o0.x = fmaxf(z[0] - tau, 0.0f);
    o0.y = fmaxf(z[1] - tau, 0.0f);
    o0.z = fmaxf(z[2] - tau, 0.0f);
    o0.w = fmaxf(z[3] - tau, 0.0f);
    o1.x = fmaxf(z[4] - tau, 0.0f);
    o1.y = fmaxf(z[5] - tau, 0.0f);
    o1.z = fmaxf(z[6] - tau, 0.0f);
    o1.w = fmaxf(z[7] - tau, 0.0f);
    v4f* q = reinterpret_cast<v4f*>(out + (size_t)row * SPX_D + lane * 8);
    __builtin_nontemporal_store(o0, q + 0);   // TH_STORE_NT: write-once stream
    __builtin_nontemporal_store(o1, q + 1);

    buf ^= 1;
  }
}

extern "C" void kernel_launch(void* const* d_in, const int* in_sizes, int n_in,
                              void* d_out, int out_size, void* d_ws, size_t ws_size,
                              hipStream_t stream) {
  const float* in = (const float*)d_in[0];
  float* out = (float*)d_out;

  const int nrows = in_sizes[0] / SPX_D;   // B = 131072 for the reference shapes
  int blocks = 2048;                       // 16384 waves -> 8 rows/wave pipeline
  const int maxb = (nrows + WAVES_PER_BLOCK - 1) / WAVES_PER_BLOCK;
  if (blocks > maxb) blocks = maxb;
  const int nwaves = blocks * WAVES_PER_BLOCK;

  sparsemax_michelot_kernel<<<blocks, WAVES_PER_BLOCK * 32, 0, stream>>>(
      in, out, nrows, nwaves);
}